// GCNEncoder_30897994727508
// MI455X (gfx1250) — compile-verified
//
#include <hip/hip_runtime.h>
#include <math.h>

// ---------------------------------------------------------------------------
// GCN encoder for MI455X (gfx1250, wave32).
//   enc1 = relu(PN(A @ (x@w1)))            [100000 x 128]
//   cat  = A @ (enc1 @ [wp|w2])            [100000 x 128]  (one fused GEMM+SpMM)
//   pool = softmax(PN(cat[:, :64]))        -> d_out[N*64 ..]
//   enc2 = relu(PN(cat[:, 64:]))           -> d_out[0 ..]
// Biases cancel under PairNorm centering and are dropped.
// ---------------------------------------------------------------------------

typedef __attribute__((ext_vector_type(2))) float v2f;
typedef __attribute__((ext_vector_type(8))) float v8f;

#define PN_EPS 1e-6f

__device__ __forceinline__ v8f wmma_f32_k4(v2f a, v2f b, v8f c) {
  // D(16x16,f32) = A(16x4,f32) x B(4x16,f32) + C   -- v_wmma_f32_16x16x4_f32
  return __builtin_amdgcn_wmma_f32_16x16x4_f32(
      /*neg_a=*/false, a, /*neg_b=*/false, b,
      /*c_mod=*/(short)0, c, /*reuse_a=*/false, /*reuse_b=*/false);
}

// C[M,128] = A[M,K] @ B[K,128];  grid.x = M/16 (M multiple of 16), block = 256.
// Wave w of the block owns output columns [16w, 16w+16).
__global__ __launch_bounds__(256) void gemm_n128_wmma(
    const float* __restrict__ A, const float* __restrict__ B,
    float* __restrict__ C, int K)
{
  __shared__ float ldsA[16 * (256 + 4)];     // padded rows: stride%256B == 16B
  const int LDA  = K + 4;
  const int row0 = blockIdx.x << 4;
  const int tid  = threadIdx.x;

  // 16xK A tile is contiguous in memory (full-K rows) -> flat float4 copy.
  const float4* srcp = (const float4*)(A + (size_t)row0 * K);
  const int K4 = K >> 2;
  for (int i = tid; i < 16 * K4; i += 256) {
    int r = i / K4;
    int c = i - r * K4;
    float4 v = srcp[i];
    float* dp = &ldsA[r * LDA + (c << 2)];
    dp[0] = v.x; dp[1] = v.y; dp[2] = v.z; dp[3] = v.w;
  }
  __syncthreads();

  const int lane = tid & 31;
  const int wav  = tid >> 5;
  const int n    = (wav << 4) + (lane & 15);  // output column
  const int m    = lane & 15;                 // A row within tile
  const int kg   = (lane >> 4) << 1;          // 0 (lanes 0-15) / 2 (lanes 16-31)

  v8f acc = {0.f, 0.f, 0.f, 0.f, 0.f, 0.f, 0.f, 0.f};
  const float* arow = &ldsA[m * LDA];
  for (int k0 = 0; k0 < K; k0 += 4) {
    v2f a, b;
    a[0] = arow[k0 + kg];
    a[1] = arow[k0 + kg + 1];
    b[0] = B[(k0 + kg) * 128 + n];
    b[1] = B[(k0 + kg + 1) * 128 + n];
    acc = wmma_f32_k4(a, b, acc);
  }

  const int mh = (lane >> 4) << 3;            // C/D rows: r (lo lanes), r+8 (hi)
  #pragma unroll
  for (int r = 0; r < 8; ++r)
    C[(size_t)(row0 + r + mh) * 128 + n] = acc[r];
}

// agg[dst, :] += w * S[src, :]   (D = 128, one wave per edge, float4 per lane)
__global__ __launch_bounds__(256) void spmm_scatter128(
    const float* __restrict__ S, const int* __restrict__ esrc,
    const int* __restrict__ edst, const float* __restrict__ ew,
    float* __restrict__ agg, int E)
{
  const int lane = threadIdx.x & 31;
  const int wave = (int)((blockIdx.x * blockDim.x + threadIdx.x) >> 5);
  const int nw   = (int)((gridDim.x * blockDim.x) >> 5);
  for (int e = wave; e < E; e += nw) {
    int en = e + nw;
    if (en < E) {  // prefetch next random gather row -> global_prefetch_b8
      __builtin_prefetch(S + (size_t)esrc[en] * 128 + (lane << 2), 0, 1);
    }
    const int   s = esrc[e];
    const int   d = edst[e];
    const float w = ew[e];
    const float4 v = ((const float4*)(S + (size_t)s * 128))[lane];
    float* p = agg + (size_t)d * 128 + (lane << 2);
    __hip_atomic_fetch_add(p + 0, w * v.x, __ATOMIC_RELAXED, __HIP_MEMORY_SCOPE_AGENT);
    __hip_atomic_fetch_add(p + 1, w * v.y, __ATOMIC_RELAXED, __HIP_MEMORY_SCOPE_AGENT);
    __hip_atomic_fetch_add(p + 2, w * v.z, __ATOMIC_RELAXED, __HIP_MEMORY_SCOPE_AGENT);
    __hip_atomic_fetch_add(p + 3, w * v.w, __ATOMIC_RELAXED, __HIP_MEMORY_SCOPE_AGENT);
  }
}

// Column sums of h[N,128] into cs[128] (for PairNorm mean).
__global__ __launch_bounds__(128) void colsum128(
    const float* __restrict__ h, float* __restrict__ cs, int nrows)
{
  const int c  = threadIdx.x;                // 0..127, coalesced across rows
  int r0 = blockIdx.x * 256;
  int r1 = r0 + 256; if (r1 > nrows) r1 = nrows;
  float s = 0.f;
  for (int r = r0; r < r1; ++r) s += h[(size_t)r * 128 + c];
  __hip_atomic_fetch_add(&cs[c], s, __ATOMIC_RELAXED, __HIP_MEMORY_SCOPE_AGENT);
}

// In-place: h = relu((h - mean) / sqrt(eps + rowssq)), D = 128, one wave/row.
__global__ __launch_bounds__(256) void pn_relu128(
    float* __restrict__ h, const float* __restrict__ cs, int nrows)
{
  const int lane = threadIdx.x & 31;
  const int row  = (int)((blockIdx.x * blockDim.x + threadIdx.x) >> 5);
  if (row >= nrows) return;
  const float invN = 1.0f / (float)nrows;
  float4 v  = ((const float4*)(h + (size_t)row * 128))[lane];
  float4 mu = ((const float4*)cs)[lane];
  v.x -= mu.x * invN; v.y -= mu.y * invN; v.z -= mu.z * invN; v.w -= mu.w * invN;
  float ss = v.x * v.x + v.y * v.y + v.z * v.z + v.w * v.w;
  #pragma unroll
  for (int m = 16; m >= 1; m >>= 1) ss += __shfl_xor(ss, m, 32);
  const float inv = 1.0f / sqrtf(PN_EPS + ss);
  v.x = fmaxf(v.x * inv, 0.f); v.y = fmaxf(v.y * inv, 0.f);
  v.z = fmaxf(v.z * inv, 0.f); v.w = fmaxf(v.w * inv, 0.f);
  ((float4*)(h + (size_t)row * 128))[lane] = v;
}

// Split epilogue: cols 0-63 -> softmax(PN(.)) -> pool; cols 64-127 -> relu(PN(.)) -> enc2.
// One wave per row; lanes 0-15 hold the pool half, 16-31 the enc2 half;
// 16-lane-group reductions (xor masks 1,2,4,8).
__global__ __launch_bounds__(256) void pn_split_out(
    const float* __restrict__ h, const float* __restrict__ cs,
    float* __restrict__ enc2, float* __restrict__ pool, int nrows)
{
  const int lane = threadIdx.x & 31;
  const int row  = (int)((blockIdx.x * blockDim.x + threadIdx.x) >> 5);
  if (row >= nrows) return;
  const float invN = 1.0f / (float)nrows;
  float4 v  = ((const float4*)(h + (size_t)row * 128))[lane];
  float4 mu = ((const float4*)cs)[lane];
  v.x -= mu.x * invN; v.y -= mu.y * invN; v.z -= mu.z * invN; v.w -= mu.w * invN;

  float ss = v.x * v.x + v.y * v.y + v.z * v.z + v.w * v.w;
  #pragma unroll
  for (int m = 8; m >= 1; m >>= 1) ss += __shfl_xor(ss, m, 32);
  const float inv = 1.0f / sqrtf(PN_EPS + ss);
  float4 u;
  u.x = v.x * inv; u.y = v.y * inv; u.z = v.z * inv; u.w = v.w * inv;

  // softmax pieces (only the lanes<16 results are consumed)
  float mx = fmaxf(fmaxf(u.x, u.y), fmaxf(u.z, u.w));
  #pragma unroll
  for (int m = 8; m >= 1; m >>= 1) mx = fmaxf(mx, __shfl_xor(mx, m, 32));
  float4 ex;
  ex.x = expf(u.x - mx); ex.y = expf(u.y - mx);
  ex.z = expf(u.z - mx); ex.w = expf(u.w - mx);
  float es = ex.x + ex.y + ex.z + ex.w;
  #pragma unroll
  for (int m = 8; m >= 1; m >>= 1) es += __shfl_xor(es, m, 32);
  const float invs = 1.0f / es;

  if (lane < 16) {
    float4 o;
    o.x = ex.x * invs; o.y = ex.y * invs; o.z = ex.z * invs; o.w = ex.w * invs;
    ((float4*)(pool + (size_t)row * 64))[lane] = o;
  } else {
    float4 o;
    o.x = fmaxf(u.x, 0.f); o.y = fmaxf(u.y, 0.f);
    o.z = fmaxf(u.z, 0.f); o.w = fmaxf(u.w, 0.f);
    ((float4*)(enc2 + (size_t)row * 64))[lane - 16] = o;
  }
}

// wcat[k, 0:64] = wp[k, :],  wcat[k, 64:128] = w2[k, :]   (k in 0..127)
__global__ void pack_wcat(const float* __restrict__ wp,
                          const float* __restrict__ w2,
                          float* __restrict__ wcat)
{
  int i = blockIdx.x * blockDim.x + threadIdx.x;
  if (i >= 128 * 128) return;
  int k = i >> 7, c = i & 127;
  wcat[i] = (c < 64) ? wp[k * 64 + c] : w2[k * 64 + (c - 64)];
}

__global__ void fill_zero(float* __restrict__ p, int n) {
  int i = blockIdx.x * blockDim.x + threadIdx.x;
  if (i < n) p[i] = 0.f;
}

extern "C" void kernel_launch(void* const* d_in, const int* in_sizes, int n_in,
                              void* d_out, int out_size, void* d_ws, size_t ws_size,
                              hipStream_t stream)
{
  const float* x    = (const float*)d_in[0];   // [N, 256]
  const int*   esrc = (const int*)d_in[1];     // [E]
  const int*   edst = (const int*)d_in[2];     // [E]
  const float* ew   = (const float*)d_in[3];   // [E]
  const float* w1   = (const float*)d_in[4];   // [256, 128]
  const float* wp   = (const float*)d_in[6];   // [128, 64]
  const float* w2   = (const float*)d_in[8];   // [128, 64]
  (void)n_in; (void)out_size; (void)ws_size;

  const int    N  = in_sizes[0] / 256;         // 100000 (multiple of 16)
  const int    E  = in_sizes[1];               // 1600000
  const size_t ND = (size_t)N * 128;

  float* bufA = (float*)d_ws;                  // support  [N,128]
  float* bufB = bufA + ND;                     // agg/enc1 [N,128]
  float* wcat = bufB + ND;                     // [128,128]
  float* cs   = wcat + 128 * 128;              // [128]

  float* out_enc2 = (float*)d_out;             // [N,64]
  float* out_pool = out_enc2 + (size_t)N * 64; // [N,64]

  const int mt  = N / 16;
  const int zgb = (int)((ND + 255) / 256);
  const int rwb = (N * 32 + 255) / 256;

  // ---- stage 1: enc1 = relu(PN(A @ (x @ w1))) ----
  gemm_n128_wmma<<<mt, 256, 0, stream>>>(x, w1, bufA, 256);
  fill_zero<<<zgb, 256, 0, stream>>>(bufB, (int)ND);
  fill_zero<<<1, 128, 0, stream>>>(cs, 128);
  spmm_scatter128<<<2048, 256, 0, stream>>>(bufA, esrc, edst, ew, bufB, E);
  colsum128<<<(N + 255) / 256, 128, 0, stream>>>(bufB, cs, N);
  pn_relu128<<<rwb, 256, 0, stream>>>(bufB, cs, N);          // bufB := enc1

  // ---- stage 2: fused [pool|enc2] = A @ (enc1 @ [wp|w2]) ----
  pack_wcat<<<64, 256, 0, stream>>>(wp, w2, wcat);
  gemm_n128_wmma<<<mt, 256, 0, stream>>>(bufB, wcat, bufA, 128);
  fill_zero<<<zgb, 256, 0, stream>>>(bufB, (int)ND);         // enc1 dead now
  fill_zero<<<1, 128, 0, stream>>>(cs, 128);
  spmm_scatter128<<<2048, 256, 0, stream>>>(bufA, esrc, edst, ew, bufB, E);
  colsum128<<<(N + 255) / 256, 128, 0, stream>>>(bufB, cs, N);
  pn_split_out<<<rwb, 256, 0, stream>>>(bufB, cs, out_enc2, out_pool, N);
}